// MultiLayerDINOv2Detector_30520037605721
// MI455X (gfx1250) — compile-verified
//
#include <hip/hip_runtime.h>

#define OUTB  7
#define CCH   256
#define HH    37
#define WW    37
#define HWSZ  (HH*WW)
#define NPROP 512
#define DFLAT (CCH*OUTB*OUTB)   // 12544
#define MROWS 2048              // B*N
#define NCLS  91

typedef __attribute__((ext_vector_type(2))) float v2f;
typedef __attribute__((ext_vector_type(8))) float v8f;

// ---------------------------------------------------------------------------
// Kernel 1: multi-level ROI align -> flat activation matrix (2048 x 12544)
// One block per roi; per-roi sample geometry precomputed once into LDS.
// Source features (22 MB total) are L2-resident on MI455X (192 MB L2).
// ---------------------------------------------------------------------------
__global__ __launch_bounds__(256) void roi_align_kernel(
    const float* __restrict__ f0, const float* __restrict__ f1,
    const float* __restrict__ f2, const float* __restrict__ f3,
    const float* __restrict__ proposals, float* __restrict__ out)
{
    const int roi = blockIdx.x;           // 0 .. 2047
    const int b   = roi / NPROP;
    const float* p = proposals + (size_t)roi * 4;
    const float px1 = p[0], py1 = p[1], px2 = p[2], py2 = p[3];
    const float bw = px2 - px1, bh = py2 - py1;
    const float area = bw * bh;
    const int lvl = (area >= 1024.f) + (area >= 4096.f) + (area >= 16384.f);
    const float* fl = (lvl == 0) ? f0 : (lvl == 1) ? f1 : (lvl == 2) ? f2 : f3;
    const float x1 = px1 - 0.5f, y1 = py1 - 0.5f;
    const float binw = bw * (1.f / OUTB), binh = bh * (1.f / OUTB);

    __shared__ int   sy0[14], sy1[14], sx0[14], sx1[14];
    __shared__ float sly[14], slx[14], svy[14], svx[14];

    const int tid = threadIdx.x;
    if (tid < 14) {
        const int   o   = tid >> 1, s = tid & 1;
        const float off = (float)o + ((float)s + 0.5f) * 0.5f;
        // y axis sample
        float y  = y1 + off * binh;
        float vy = (y >= -1.f && y <= (float)HH) ? 1.f : 0.f;
        float yc = fminf(fmaxf(y, 0.f), (float)(HH - 1));
        float y0f = floorf(yc);
        int   y0  = (int)y0f;
        sy0[tid] = y0;
        sy1[tid] = min(y0 + 1, HH - 1);
        sly[tid] = yc - y0f;
        svy[tid] = vy;
        // x axis sample
        float x  = x1 + off * binw;
        float vx = (x >= -1.f && x <= (float)WW) ? 1.f : 0.f;
        float xc = fminf(fmaxf(x, 0.f), (float)(WW - 1));
        float x0f = floorf(xc);
        int   x0  = (int)x0f;
        sx0[tid] = x0;
        sx1[tid] = min(x0 + 1, WW - 1);
        slx[tid] = xc - x0f;
        svx[tid] = vx;
    }
    __syncthreads();

    const float* fb   = fl + (size_t)b * CCH * HWSZ;
    float*       orow = out + (size_t)roi * DFLAT;

    for (int idx = tid; idx < DFLAT; idx += 256) {
        const int c   = idx / 49;
        const int bin = idx - c * 49;
        const int oy  = bin / 7, ox = bin - oy * 7;
        const float* fc = fb + (size_t)c * HWSZ;
        float acc = 0.f;
        #pragma unroll
        for (int sy = 0; sy < 2; ++sy) {
            const int   iy = oy * 2 + sy;
            const int   yy0 = sy0[iy], yy1 = sy1[iy];
            const float ly = sly[iy], hy = 1.f - ly, vy = svy[iy];
            #pragma unroll
            for (int sx = 0; sx < 2; ++sx) {
                const int   ix = ox * 2 + sx;
                const int   xx0 = sx0[ix], xx1 = sx1[ix];
                const float lx = slx[ix], hx = 1.f - lx;
                const float v  = vy * svx[ix];
                const float t = hy * hx * fc[yy0 * WW + xx0]
                              + hy * lx * fc[yy0 * WW + xx1]
                              + ly * hx * fc[yy1 * WW + xx0]
                              + ly * lx * fc[yy1 * WW + xx1];
                acc += v * t;
            }
        }
        orow[idx] = acc * 0.25f;
    }
}

// ---------------------------------------------------------------------------
// Kernel 2: generic fp32 WMMA GEMM  C = act(A @ B + bias)
// A: MxK row-major, B: KxN row-major, C: MxN row-major. K % 16 == 0, M % 128 == 0.
// N is a COMPILE-TIME template parameter so all in-loop B offsets become
// 24-bit immediate IOFFSETs (one scalar pointer bump per unrolled block),
// matching the A-side codegen. Block = 256 threads = 8 waves over a 128x64
// C tile; each wave owns 32x32 via 2x2 fragments of v_wmma_f32_16x16x4_f32
// (exact fp32 math; these GEMMs are bandwidth bound on MI455X, so the small-K
// f32 WMMA shape costs nothing vs bf16 while matching the reference exactly).
// Edge handling: B column indices CLAMPED to N-1 (column n of B only affects
// column n of C, so OOB lanes redundantly compute column N-1); stores guarded.
// EXEC stays all-ones through the whole K loop (WMMA requirement).
// ---------------------------------------------------------------------------
template<int N, int RELU>
__global__ __launch_bounds__(256) void gemm_bias_act_wmma(
    const float* __restrict__ A, const float* __restrict__ B,
    const float* __restrict__ bias, float* __restrict__ C, int K)
{
    const int lane = threadIdx.x & 31;
    const int wave = threadIdx.x >> 5;   // 0..7
    const int ml   = lane & 15;          // matrix row (A) / col (B,C) within tile
    const int kg   = lane >> 4;          // K-group (A/B) and M-group (C)

    const int m0 = blockIdx.y * 128 + (wave & 3) * 32;
    const int n0 = blockIdx.x * 64 + (wave >> 2) * 32;

    const int col0 = n0 + ml;
    const int col1 = n0 + 16 + ml;
    const int c0   = (col0 < N) ? col0 : (N - 1);   // clamped (always addressable)
    const int c1   = (col1 < N) ? col1 : (N - 1);

    // Per-lane streaming pointers; all lane-dependent terms folded into the
    // base so in-loop offsets are uniform compile-time constants.
    const float* Ap0 = A + (size_t)(m0 + ml) * K + 2 * kg;
    const float* Ap1 = Ap0 + (size_t)16 * K;
    const float* Bp0 = B + (size_t)(2 * kg) * N + c0;
    const float* Bp1 = B + (size_t)(2 * kg) * N + c1;

    v8f c00 = {}, c01 = {}, c10 = {}, c11 = {};

    #pragma unroll 4
    for (int k0 = 0; k0 < K; k0 += 4) {
        v2f a0 = *(const v2f*)Ap0;       // (m0+ml,    k+2kg..+1) : global_load_b64
        v2f a1 = *(const v2f*)Ap1;       // (m0+16+ml, k+2kg..+1)
        v2f b0, b1;
        b0.x = Bp0[0]; b0.y = Bp0[N];    // (k+2kg, c0), (k+2kg+1, c0)
        b1.x = Bp1[0]; b1.y = Bp1[N];
        Ap0 += 4; Ap1 += 4;
        Bp0 += 4 * N; Bp1 += 4 * N;

        c00 = __builtin_amdgcn_wmma_f32_16x16x4_f32(false, a0, false, b0, (short)0, c00, false, false);
        c01 = __builtin_amdgcn_wmma_f32_16x16x4_f32(false, a0, false, b1, (short)0, c01, false, false);
        c10 = __builtin_amdgcn_wmma_f32_16x16x4_f32(false, a1, false, b0, (short)0, c10, false, false);
        c11 = __builtin_amdgcn_wmma_f32_16x16x4_f32(false, a1, false, b1, (short)0, c11, false, false);
    }

    const bool  v0    = col0 < N;
    const bool  v1    = col1 < N;
    const float bias0 = bias[c0];
    const float bias1 = bias[c1];

    #pragma unroll
    for (int i = 0; i < 8; ++i) {
        const int r0 = m0 + i + 8 * kg;   // first fragment row block
        const int r1 = r0 + 16;           // second fragment row block
        if (v0) {
            float s = c00[i] + bias0; if (RELU) s = fmaxf(s, 0.f);
            C[(size_t)r0 * N + col0] = s;
            s = c10[i] + bias0; if (RELU) s = fmaxf(s, 0.f);
            C[(size_t)r1 * N + col0] = s;
        }
        if (v1) {
            float s = c01[i] + bias1; if (RELU) s = fmaxf(s, 0.f);
            C[(size_t)r0 * N + col1] = s;
            s = c11[i] + bias1; if (RELU) s = fmaxf(s, 0.f);
            C[(size_t)r1 * N + col1] = s;
        }
    }
}

// ---------------------------------------------------------------------------
// Launch
// ---------------------------------------------------------------------------
extern "C" void kernel_launch(void* const* d_in, const int* in_sizes, int n_in,
                              void* d_out, int out_size, void* d_ws, size_t ws_size,
                              hipStream_t stream) {
    (void)in_sizes; (void)n_in; (void)out_size; (void)ws_size;

    const float* f0        = (const float*)d_in[0];
    const float* f1        = (const float*)d_in[1];
    const float* f2        = (const float*)d_in[2];
    const float* f3        = (const float*)d_in[3];
    const float* proposals = (const float*)d_in[4];
    const float* w1 = (const float*)d_in[5];
    const float* b1 = (const float*)d_in[6];
    const float* w2 = (const float*)d_in[7];
    const float* b2 = (const float*)d_in[8];
    const float* wc = (const float*)d_in[9];
    const float* bc = (const float*)d_in[10];
    const float* wb = (const float*)d_in[11];
    const float* bb = (const float*)d_in[12];

    float* scores = (float*)d_out;                           // 2048 x 91
    float* bbox   = (float*)d_out + (size_t)MROWS * NCLS;    // 2048 x 4

    char*  ws   = (char*)d_ws;
    float* flat = (float*)ws;                                           // 2048 x 12544 (98 MB)
    float* h1   = (float*)(ws + (size_t)MROWS * DFLAT * sizeof(float)); // 2048 x 1024
    float* h2   = (float*)ws;  // alias: flat is dead once h1 is produced

    // 1) ROI align -> flat
    roi_align_kernel<<<MROWS, 256, 0, stream>>>(f0, f1, f2, f3, proposals, flat);

    // 2) h1 = relu(flat @ w1 + b1)   M=2048 N=1024 K=12544
    dim3 g1(1024 / 64, MROWS / 128);
    gemm_bias_act_wmma<1024, 1><<<g1, 256, 0, stream>>>(flat, w1, b1, h1, DFLAT);

    // 3) h2 = relu(h1 @ w2 + b2)     M=2048 N=1024 K=1024
    gemm_bias_act_wmma<1024, 1><<<g1, 256, 0, stream>>>(h1, w2, b2, h2, 1024);

    // 4) scores = h2 @ wc + bc       M=2048 N=91  K=1024
    dim3 g3((NCLS + 63) / 64, MROWS / 128);
    gemm_bias_act_wmma<NCLS, 0><<<g3, 256, 0, stream>>>(h2, wc, bc, scores, 1024);

    // 5) bbox  = h2 @ wb + bb        M=2048 N=4   K=1024
    dim3 g4(1, MROWS / 128);
    gemm_bias_act_wmma<4, 0><<<g4, 256, 0, stream>>>(h2, wb, bb, bbox, 1024);
}